// TA_32762010534459
// MI455X (gfx1250) — compile-verified
//
#include <hip/hip_runtime.h>
#include <hip/hip_bf16.h>

// Problem constants (from reference)
#define NT_TOT 128          // N*T
#define C_CH   256          // C_IN
#define HH     28
#define WW     28
#define HW     784          // 28*28
#define WP     30           // padded width  (1-px halo)
#define HWP    900          // padded plane  30*30
#define C_RED  16
#define K_TOT  2304         // C_CH * 9 (implicit-GEMM K)

typedef float v2f __attribute__((ext_vector_type(2)));
typedef float v8f __attribute__((ext_vector_type(8)));

// ---------------------------------------------------------------------------
// Kernel 1: temporal shift (depthwise k=3 over segment axis) + spatial mean.
// Writes x_shift into a zero-padded 30x30 plane so the conv's 9 im2col taps
// are always in-bounds (constant immediate offsets, no masks).
// ---------------------------------------------------------------------------
__global__ __launch_bounds__(256) void ta_shift_pool_kernel(
    const float* __restrict__ x, const float* __restrict__ taw,
    float* __restrict__ xs, float* __restrict__ pooled)
{
    const int bc = blockIdx.x;       // nt*256 + c
    const int nt = bc >> 8;
    const int c  = bc & 255;
    const int t  = nt & 7;

    const float wa = taw[c * 3 + 0];
    const float wb = taw[c * 3 + 1];
    const float wc = taw[c * 3 + 2];

    const size_t baseIn  = (size_t)bc * HW;
    const size_t baseOut = (size_t)bc * HWP;
    const size_t stride  = (size_t)C_CH * HW;
    const bool hasP = (t > 0);
    const bool hasN = (t < 7);

    float sum = 0.f;
    for (int p = threadIdx.x; p < HWP; p += 256) {
        const int h2 = p / WP;
        const int w2 = p - h2 * WP;
        float v = 0.f;
        if (h2 >= 1 && h2 <= HH && w2 >= 1 && w2 <= WW) {
            const int q = (h2 - 1) * WW + (w2 - 1);
            v = wb * x[baseIn + q];
            if (hasP) v += wa * x[baseIn - stride + q];
            if (hasN) v += wc * x[baseIn + stride + q];
            sum += v;
        }
        xs[baseOut + p] = v;
    }

    __shared__ float red[256];
    red[threadIdx.x] = sum;
    __syncthreads();
    #pragma unroll
    for (int s = 128; s > 0; s >>= 1) {
        if (threadIdx.x < (unsigned)s) red[threadIdx.x] += red[threadIdx.x + s];
        __syncthreads();
    }
    if (threadIdx.x == 0) pooled[bc] = red[0] * (1.f / (float)HW);
}

// ---------------------------------------------------------------------------
// Kernel 2: full SE path in one workgroup (tiny: ~2M MACs).
// ---------------------------------------------------------------------------
__global__ __launch_bounds__(256) void se_gate_kernel(
    const float* __restrict__ pooled, const float* __restrict__ cw,
    const float* __restrict__ gamma, const float* __restrict__ beta,
    const float* __restrict__ w1x1, float* __restrict__ gate)
{
    __shared__ float yc[C_RED * 128];   // [j][b*8+t]
    __shared__ float sc[C_RED], sh[C_RED];
    const int tid = threadIdx.x;

    for (int idx = tid; idx < 2048; idx += 256) {
        const int b = idx >> 7;
        const int j = (idx >> 3) & 15;
        const int t = idx & 7;
        const float* Y = pooled + b * 2048;   // [c2*8 + tt]
        const float* W = cw + j * (C_CH * 3);
        float acc = 0.f;
        for (int c2 = 0; c2 < C_CH; ++c2) {
            const float* y8 = Y + c2 * 8;
            const float* w3 = W + c2 * 3;
            if (t > 0) acc += w3[0] * y8[t - 1];
            acc += w3[1] * y8[t];
            if (t < 7) acc += w3[2] * y8[t + 1];
        }
        yc[j * 128 + b * 8 + t] = acc;
    }
    __syncthreads();

    if (tid < C_RED) {
        float s = 0.f, s2 = 0.f;
        for (int i = 0; i < 128; ++i) {
            float v = yc[tid * 128 + i];
            s += v; s2 += v * v;
        }
        const float mean = s * (1.f / 128.f);
        const float var  = s2 * (1.f / 128.f) - mean * mean;
        const float inv  = rsqrtf(var + 1e-5f);
        const float g    = gamma[tid];
        sc[tid] = g * inv;
        sh[tid] = beta[tid] - mean * g * inv;
    }
    __syncthreads();

    for (int idx = tid; idx < NT_TOT * C_CH; idx += 256) {
        const int b = idx >> 11;
        const int o = (idx >> 3) & 255;
        const int t = idx & 7;
        const float* Wo = w1x1 + o * C_RED;
        float acc = 0.f;
        #pragma unroll
        for (int j = 0; j < C_RED; ++j) {
            float v = yc[j * 128 + b * 8 + t] * sc[j] + sh[j];
            v = fmaxf(v, 0.f);
            acc += Wo[j] * v;
        }
        gate[idx] = 1.f / (1.f + __expf(-acc));
    }
}

// ---------------------------------------------------------------------------
// Kernel 3: 3x3 conv as implicit GEMM on V_WMMA_F32_16X16X4_F32.
//   M=256, N=100352 pixels, K=2304. Block: 8 waves -> 128x64 tile, each wave
//   32x32 (4 accumulators). K chunked by 36 (4 channels x 9 taps).
//   A-staging: thread = half-row of weights (18 contiguous floats).
//   B-staging: thread = (pixel, channel) -> 9 taps at constant immediate
//   offsets in the padded plane, pre-scaled by its (1+gate).
//   Bs stored [N][K] so B fragments are contiguous pairs (no repack moves).
//   Double-buffered LDS, register prefetch, 1 barrier per chunk.
// ---------------------------------------------------------------------------
#define BM   128
#define BN   64
#define KC   36
#define NCHUNK (K_TOT / KC)   // 64
#define LDA2 38               // As row stride (8B-aligned rows, no read conflicts)
#define LDBK 37               // Bs row stride ([n][k] layout)
#define AE   18               // A floats per thread: 128*36/256
#define BE   9                // B floats per thread:  36*64/256

__global__ __launch_bounds__(256) void conv3x3_wmma_kernel(
    const float* __restrict__ xs, const float* __restrict__ gate,
    const float* __restrict__ wgt, const float* __restrict__ bias,
    float* __restrict__ out)
{
    __shared__ float As[2][BM * LDA2];   // 2*4864*4 = 38,912 B
    __shared__ float Bs[2][BN * LDBK];   // 2*2368*4 = 18,944 B

    const int tid   = threadIdx.x;
    const int lane  = tid & 31;
    const int wave  = tid >> 5;
    const int mOff  = (wave >> 1) * 32;     // 0,32,64,96
    const int nOff  = (wave & 1) * 32;      // 0,32
    const int mBase = blockIdx.y * BM;
    const int nBase = blockIdx.x * BN;
    const int lhalf = lane >> 4;            // K pair select (ISA frag layout)
    const int l15   = lane & 15;

    // ---- A staging map: row = tid>>1, cols [half*18, half*18+18) ----
    const int ai  = tid >> 1;
    const int acj = (tid & 1) * AE;
    const float* aSrc = wgt + (size_t)(mBase + ai) * K_TOT + acj;  // +KC/chunk
    const int aLdsOff = ai * LDA2 + acj;

    // ---- B staging map: pixel n = tid>>2, channel-in-chunk ci0 = tid&3 ----
    const int n   = tid >> 2;
    const int ci0 = tid & 3;
    const int pix = nBase + n;
    const int nt  = pix / HW;
    const int rem = pix - nt * HW;
    const int h   = rem / WW;
    const int w   = rem - h * WW;
    const float* bSrc = xs + (size_t)(nt * C_CH + ci0) * HWP
                           + (h + 1) * WP + (w + 1);               // +4*HWP/chunk
    int gOff = nt * C_CH + ci0;                                    // +4/chunk
    const int bLdsOff = n * LDBK + ci0 * BE;

    float aReg[AE], bReg[BE];
    auto loadRegs = [&]() {
        #pragma unroll
        for (int e = 0; e < AE; ++e) aReg[e] = aSrc[e];
        aSrc += KC;
        const float gmul = 1.f + gate[gOff];
        gOff += 4;
        // 9 taps: dy,dx in {-1,0,1} -> constant offsets in padded plane
        #pragma unroll
        for (int dy = 0; dy < 3; ++dy)
            #pragma unroll
            for (int dx = 0; dx < 3; ++dx)
                bReg[dy * 3 + dx] = bSrc[(dy - 1) * WP + (dx - 1)] * gmul;
        bSrc += 4 * HWP;
    };
    auto storeRegs = [&](float* Ad, float* Bd) {
        #pragma unroll
        for (int e = 0; e < AE; ++e) Ad[aLdsOff + e] = aReg[e];
        #pragma unroll
        for (int e = 0; e < BE; ++e) Bd[bLdsOff + e] = bReg[e];
    };

    v8f acc00 = {}, acc01 = {}, acc10 = {}, acc11 = {};

    auto compute = [&](const float* Ab, const float* Bb) {
        const float* a0p = Ab + (mOff + l15) * LDA2 + 2 * lhalf;
        const float* a1p = a0p + 16 * LDA2;
        const float* b0p = Bb + (nOff + l15) * LDBK + 2 * lhalf;
        const float* b1p = b0p + 16 * LDBK;
        #pragma unroll
        for (int kk = 0; kk < KC; kk += 4) {
            v2f a0, a1, b0, b1;
            a0.x = a0p[kk]; a0.y = a0p[kk + 1];
            a1.x = a1p[kk]; a1.y = a1p[kk + 1];
            b0.x = b0p[kk]; b0.y = b0p[kk + 1];
            b1.x = b1p[kk]; b1.y = b1p[kk + 1];
            acc00 = __builtin_amdgcn_wmma_f32_16x16x4_f32(false, a0, false, b0, (short)0, acc00, false, false);
            acc01 = __builtin_amdgcn_wmma_f32_16x16x4_f32(false, a0, false, b1, (short)0, acc01, false, false);
            acc10 = __builtin_amdgcn_wmma_f32_16x16x4_f32(false, a1, false, b0, (short)0, acc10, false, false);
            acc11 = __builtin_amdgcn_wmma_f32_16x16x4_f32(false, a1, false, b1, (short)0, acc11, false, false);
        }
    };

    // ---- prologue: fill buffer 0 ----
    loadRegs();
    storeRegs(As[0], Bs[0]);
    __syncthreads();

    // ---- main loop ----
    for (int c = 0; c < NCHUNK; ++c) {
        const int cur = c & 1;
        if (c + 1 < NCHUNK) loadRegs();
        compute(As[cur], Bs[cur]);
        if (c + 1 < NCHUNK) storeRegs(As[cur ^ 1], Bs[cur ^ 1]);
        __syncthreads();
    }

    // ---- epilogue: D tiles -> NCHW + bias ----
    #pragma unroll
    for (int ni = 0; ni < 2; ++ni) {
        const int opix = nBase + nOff + ni * 16 + l15;
        const int ont  = opix / HW;
        const int orem = opix - ont * HW;
        const size_t outBase = (size_t)ont * C_CH * HW + orem;
        const v8f am0 = ni ? acc01 : acc00;
        const v8f am1 = ni ? acc11 : acc10;
        #pragma unroll
        for (int v = 0; v < 8; ++v) {
            const int m0 = mBase + mOff + v + 8 * lhalf;
            const int m1 = m0 + 16;
            out[outBase + (size_t)m0 * HW] = am0[v] + bias[m0];
            out[outBase + (size_t)m1 * HW] = am1[v] + bias[m1];
        }
    }
}

// ---------------------------------------------------------------------------
extern "C" void kernel_launch(void* const* d_in, const int* in_sizes, int n_in,
                              void* d_out, int out_size, void* d_ws, size_t ws_size,
                              hipStream_t stream) {
    const float* x     = (const float*)d_in[0];   // [128,256,28,28]
    const float* taw   = (const float*)d_in[1];   // [256,3]
    const float* cw    = (const float*)d_in[2];   // [16,256,3]
    const float* gamma = (const float*)d_in[3];   // [16]
    const float* beta  = (const float*)d_in[4];   // [16]
    const float* w1x1  = (const float*)d_in[5];   // [256,16]
    const float* wgt   = (const float*)d_in[6];   // [256,256,3,3] == A[256][2304]
    const float* bias  = (const float*)d_in[7];   // [256]
    float* out = (float*)d_out;

    float* xs     = (float*)d_ws;                        // 128*256*900 f32 (padded)
    float* pooled = xs + (size_t)NT_TOT * C_CH * HWP;    // 32768 f32
    float* gate   = pooled + NT_TOT * C_CH;              // 32768 f32

    ta_shift_pool_kernel<<<NT_TOT * C_CH, 256, 0, stream>>>(x, taw, xs, pooled);
    se_gate_kernel<<<1, 256, 0, stream>>>(pooled, cw, gamma, beta, w1x1, gate);

    dim3 grid((NT_TOT * HW) / BN, C_CH / BM);   // 1568 x 2
    conv3x3_wmma_kernel<<<grid, 256, 0, stream>>>(xs, gate, wgt, bias, out);
}